// Model_45002667327785
// MI455X (gfx1250) — compile-verified
//
#include <hip/hip_runtime.h>

typedef float v4f __attribute__((ext_vector_type(4)));
typedef int   v4i __attribute__((ext_vector_type(4)));

#ifndef DMID
#define DMID 64
#endif
#define ROWS_PER_THREAD 8

// out[row, 0:64] = relu(W[x[row], 0:64] + b[0:64]) + 1e-5
//
// Each thread owns one float4 column chunk (c4 = tid&15) of 8 CONSECUTIVE
// rows, so its 8 indices load as two b128 vector loads (32B-aligned).
// Per iteration a wave stores two contiguous 256 B row segments; a block
// covers 128 consecutive rows (32 KB contiguous output span).
//
// NT stores keep the 1.07 GB output stream from evicting the 25.6 MB
// embedding table out of the 192 MB L2; HBM traffic ~= output + indices
// => ~47 us floor @ 23.3 TB/s.
__global__ __launch_bounds__(256) void embed_bias_relu_fast(
    const int* __restrict__ x,      // [n_rows] int32 indices
    const float* __restrict__ W,    // [NUM_CATEGORIES, DMID] f32
    const float* __restrict__ b,    // [DMID] f32
    v4f* __restrict__ out)          // [n_rows * DMID/4] float4 chunks
{
    const unsigned tid = blockIdx.x * 256u + threadIdx.x;
    const unsigned c4  = tid & 15u;                 // float4 column chunk
    const unsigned r0  = (tid >> 4) * ROWS_PER_THREAD; // first row of group

    // Two 128-bit index loads (8 consecutive int32 indices, 32B-aligned).
    const v4i ia = *(const v4i*)(x + r0);
    const v4i ib = *(const v4i*)(x + r0 + 4);
    int idx[ROWS_PER_THREAD] = { ia.x, ia.y, ia.z, ia.w,
                                 ib.x, ib.y, ib.z, ib.w };

    const v4f bb = ((const v4f*)b)[c4];             // bias chunk (cache-hot)
    const unsigned obase = r0 * (DMID / 4) + c4;    // first float4 slot

    #pragma unroll
    for (int i = 0; i < ROWS_PER_THREAD; ++i) {
        const unsigned woff = (unsigned)idx[i] * (unsigned)DMID + c4 * 4u;
        const v4f w = *(const v4f*)(W + woff);      // L2-resident gather
        const v4f s = w + bb;                       // f32 adds, matches ref
        v4f r;
        r.x = fmaxf(s.x, 0.0f) + 1e-5f;
        r.y = fmaxf(s.y, 0.0f) + 1e-5f;
        r.z = fmaxf(s.z, 0.0f) + 1e-5f;
        r.w = fmaxf(s.w, 0.0f) + 1e-5f;
        __builtin_nontemporal_store(r, out + (obase + (unsigned)i * (DMID / 4)));
    }
}

// Guarded fallback for shapes that don't divide (not used for 2048x2048x64).
__global__ __launch_bounds__(256) void embed_bias_relu_guarded(
    const int* __restrict__ x,
    const float* __restrict__ W,
    const float* __restrict__ b,
    v4f* __restrict__ out,
    unsigned total4)
{
    const unsigned tid = blockIdx.x * 256u + threadIdx.x;
    const unsigned c4  = tid & 15u;
    const unsigned r0  = (tid >> 4) * ROWS_PER_THREAD;
    const v4f bb = ((const v4f*)b)[c4];

    #pragma unroll
    for (int i = 0; i < ROWS_PER_THREAD; ++i) {
        const unsigned gid = (r0 + (unsigned)i) * (DMID / 4) + c4;
        if (gid < total4) {
            const int idx = x[r0 + (unsigned)i];
            const unsigned woff = (unsigned)idx * (unsigned)DMID + c4 * 4u;
            const v4f w = *(const v4f*)(W + woff);
            const v4f s = w + bb;
            v4f r;
            r.x = fmaxf(s.x, 0.0f) + 1e-5f;
            r.y = fmaxf(s.y, 0.0f) + 1e-5f;
            r.z = fmaxf(s.z, 0.0f) + 1e-5f;
            r.w = fmaxf(s.w, 0.0f) + 1e-5f;
            __builtin_nontemporal_store(r, out + gid);
        }
    }
}

extern "C" void kernel_launch(void* const* d_in, const int* in_sizes, int n_in,
                              void* d_out, int out_size, void* d_ws, size_t ws_size,
                              hipStream_t stream) {
    const int*   x = (const int*)  d_in[0];   // [2048*2048] int32
    const float* W = (const float*)d_in[1];   // [100000, 64] f32
    const float* b = (const float*)d_in[2];   // [64] f32
    v4f* out = (v4f*)d_out;                   // [2048*2048*16] float4

    const unsigned long long n_rows = (unsigned long long)in_sizes[0]; // 4,194,304
    const unsigned long long total4 = n_rows * (DMID / 4);             // 67,108,864

    const int block = 256;
    // One thread per (row-group of 8) x (float4 column of 16)
    const unsigned long long threads_needed =
        ((n_rows + ROWS_PER_THREAD - 1) / ROWS_PER_THREAD) * 16ull;    // 8,388,608
    const unsigned grid = (unsigned)((threads_needed + block - 1) / block); // 32,768

    if ((n_rows % ROWS_PER_THREAD == 0) &&
        ((unsigned long long)grid * block == threads_needed)) {
        embed_bias_relu_fast<<<grid, block, 0, stream>>>(x, W, b, out);
    } else {
        embed_bias_relu_guarded<<<grid, block, 0, stream>>>(
            x, W, b, out, (unsigned)total4);
    }
}